// Lossfunction_60859686584602
// MI455X (gfx1250) — compile-verified
//
#include <hip/hip_runtime.h>
#include <hip/hip_bf16.h>
#include <math.h>
#include <stdint.h>

// Problem constants (match reference)
#define NBATCH   1024
#define ROWS     64        // P
#define DIMS     1024      // D
#define NUM_POS  8
#define NUM_NEG  55
#define BK       64        // K-chunk staged in LDS
#define NCHUNK   (DIMS / BK)
#define LSTRIDE  68        // padded LDS row stride (floats): 16B-aligned, bank-spread

typedef __attribute__((ext_vector_type(2))) float v2f;
typedef __attribute__((ext_vector_type(8))) float v8f;

// Issue the async DMA (GLOBAL_LOAD_ASYNC_TO_LDS_B128) of one 64xBK fp32 chunk
// into an LDS buffer: 8 B128 lane-transfers per thread, tracked by ASYNCcnt.
__device__ __forceinline__ void issue_chunk_dma(float* __restrict__ buf,
                                                uint64_t xb_base, int t, int kb) {
#pragma unroll
    for (int i = 0; i < 8; ++i) {
        const int flat = t + i * 128;
        const int r    = flat >> 4;      // 0..63
        const int cq   = flat & 15;      // float4 column group
        const uint32_t goff = (uint32_t)((r * DIMS + kb + cq * 4) * sizeof(float));
        const uint32_t loff = (uint32_t)(uintptr_t)(&buf[r * LSTRIDE + cq * 4]);
        asm volatile("global_load_async_to_lds_b128 %0, %1, %2"
                     :: "v"(loff), "v"(goff), "s"(xb_base)
                     : "memory");
    }
}

// One workgroup (4 wave32) per batch: double-buffered async DMA into LDS
// overlapped with the 64x64 fp32 Gram computed via V_WMMA_F32_16X16X4_F32,
// then the NCE logsumexp epilogue.
__global__ __launch_bounds__(128) void nce_gram_kernel(const float* __restrict__ x,
                                                       float* __restrict__ partial) {
    __shared__ float lds[2][ROWS * LSTRIDE];   // double-buffered X chunks; [0] reused for G
    __shared__ float nrm[ROWS];
    __shared__ float plos[NUM_POS];

    const int t    = threadIdx.x;
    const int b    = blockIdx.x;
    const int wave = t >> 5;
    const int lane = t & 31;
    const int l16  = lane & 15;
    const int hi   = lane >> 4;          // 0 for lanes 0-15, 1 for lanes 16-31
    const int m0   = wave * 16;          // this wave's M tile origin

    const float* __restrict__ xb = x + (size_t)b * (ROWS * DIMS);
    const uint64_t xb_base = (uint64_t)(uintptr_t)xb;   // uniform -> SGPR pair

    v8f acc[4];
#pragma unroll
    for (int i = 0; i < 4; ++i)
        acc[i] = (v8f){0.f, 0.f, 0.f, 0.f, 0.f, 0.f, 0.f, 0.f};

    // Prologue: start DMA of chunk 0 into buffer 0.
    issue_chunk_dma(&lds[0][0], xb_base, t, 0);

    for (int c = 0; c < NCHUNK; ++c) {
        const int cur = c & 1;

        if (c + 1 < NCHUNK) {
            // Kick off chunk c+1 into the other buffer, then wait only for the
            // previous 8 transfers (counted ASYNCcnt wait): next chunk's DMA
            // streams while the matrix pipe consumes the current one.
            issue_chunk_dma(&lds[cur ^ 1][0], xb_base, t, (c + 1) * BK);
            asm volatile("s_wait_asynccnt 0x8" ::: "memory");
        } else {
            asm volatile("s_wait_asynccnt 0x0" ::: "memory");
        }
        __syncthreads();

        const float* __restrict__ buf = &lds[cur][0];

        // 16 WMMA K-steps over this chunk; 4 N-tiles per wave.
#pragma unroll
        for (int kk = 0; kk < BK; kk += 4) {
            const int kc = kk + 2 * hi;  // K position for this lane half (VGPR0/1 = K, K+1)
            const v2f a = *reinterpret_cast<const v2f*>(&buf[(m0 + l16) * LSTRIDE + kc]);
#pragma unroll
            for (int tn = 0; tn < 4; ++tn) {
                const v2f bf = *reinterpret_cast<const v2f*>(&buf[(tn * 16 + l16) * LSTRIDE + kc]);
                acc[tn] = __builtin_amdgcn_wmma_f32_16x16x4_f32(
                    /*neg_a=*/false, a, /*neg_b=*/false, bf,
                    /*c_mod=*/(short)0, acc[tn],
                    /*reuse_a=*/false, /*reuse_b=*/false);
            }
        }
        // DScnt is flushed by the barrier, so the buffer consumed here may be
        // safely refilled by the DMA issued in iteration c+1.
        __syncthreads();
    }

    // Spill Gram G (64x64) into buffer 0.
    // C layout: VGPR v -> M = v + 8*hi, lane&15 -> N.
    float* __restrict__ G = &lds[0][0];
#pragma unroll
    for (int tn = 0; tn < 4; ++tn) {
#pragma unroll
        for (int v = 0; v < 8; ++v) {
            G[(m0 + v + 8 * hi) * LSTRIDE + tn * 16 + l16] = acc[tn][v];
        }
    }
    __syncthreads();

    // Row norms from the diagonal.
    if (t < ROWS) nrm[t] = sqrtf(G[t * LSTRIDE + t]);
    __syncthreads();

    // NCE loss per positive p: logits = [cos_pos, cos_neg_0..54] / TAU
    if (t < NUM_POS) {
        const int   pr      = 1 + t;       // pos row index
        const float inv_tau = 10.0f;       // 1 / 0.1
        const float np      = nrm[pr];

        const float g0 = G[0 * LSTRIDE + pr];                       // orig . pos_p
        const float l0 = (g0 / fmaxf(nrm[0] * np, 1e-6f)) * inv_tau;

        float m = l0;
        for (int n = 0; n < NUM_NEG; ++n) {
            const float g  = G[pr * LSTRIDE + 9 + n];               // pos_p . neg_n
            const float ln = (g / fmaxf(np * nrm[9 + n], 1e-6f)) * inv_tau;
            m = fmaxf(m, ln);
        }
        float s = expf(l0 - m);
        for (int n = 0; n < NUM_NEG; ++n) {
            const float g  = G[pr * LSTRIDE + 9 + n];
            const float ln = (g / fmaxf(np * nrm[9 + n], 1e-6f)) * inv_tau;
            s += expf(ln - m);
        }
        plos[t] = m + logf(s) - l0;        // logsumexp - logit0
    }
    __syncthreads();

    if (t == 0) {
        float s = 0.f;
#pragma unroll
        for (int i = 0; i < NUM_POS; ++i) s += plos[i];
        partial[b] = s;
    }
}

// Deterministic fixed-order reduction of 1024 per-batch partials -> mean loss.
__global__ __launch_bounds__(256) void nce_reduce_kernel(const float* __restrict__ partial,
                                                         float* __restrict__ out) {
    __shared__ float sm[256];
    const int t = threadIdx.x;
    float v = partial[t] + partial[t + 256] + partial[t + 512] + partial[t + 768];
    sm[t] = v;
    __syncthreads();
    for (int off = 128; off > 0; off >>= 1) {
        if (t < off) sm[t] += sm[t + off];
        __syncthreads();
    }
    if (t == 0) out[0] = sm[0] * (1.0f / (float)(NBATCH * NUM_POS));
}

extern "C" void kernel_launch(void* const* d_in, const int* in_sizes, int n_in,
                              void* d_out, int out_size, void* d_ws, size_t ws_size,
                              hipStream_t stream) {
    const float* x       = (const float*)d_in[0];   // (B*P, D) fp32; labels (d_in[1]) unused
    float*       out     = (float*)d_out;           // scalar
    float*       partial = (float*)d_ws;            // NBATCH floats of scratch

    nce_gram_kernel<<<NBATCH, 128, 0, stream>>>(x, partial);
    nce_reduce_kernel<<<1, 256, 0, stream>>>(partial, out);
}